// EquivariantProductBasisBlock_42125039239737
// MI455X (gfx1250) — compile-verified
//
#include <hip/hip_runtime.h>

typedef __attribute__((ext_vector_type(2))) float v2f;
typedef __attribute__((ext_vector_type(8))) float v8f;

#define NNODES 2048
#define NCHAN  64
#define NI     16
#define NELEM  10
#define RTOT   1024   // 256 (t0: m=0) + 768 (t1: m=1..3), r = m*256 + i*16 + j
#define MAX_TILES 144

// ---------------------------------------------------------------------------
// Kernel A3: Ueff3, stored interleaved for b64 A-operand loads:
//   layout idx = ((e*64+c)*8 + lp) * 2048 + r*2 + par,  where l = 2*lp + par
//   value = sum_k U3_t[m,i,j,l,k] * W3_t[e,k,c]
// ---------------------------------------------------------------------------
__global__ void build_ueff3(const float* __restrict__ U3_0, const float* __restrict__ W3_0,
                            const float* __restrict__ U3_1, const float* __restrict__ W3_1,
                            float* __restrict__ Ueff3) {
    int idx = blockIdx.x * blockDim.x + threadIdx.x;
    if (idx >= NELEM * NCHAN * NI * RTOT) return;
    int par = idx & 1;
    int r   = (idx >> 1) & (RTOT - 1);
    int lp  = (idx >> 11) & 7;
    int ec  = idx >> 14;           // e*64 + c
    int c   = ec & 63;
    int e   = ec >> 6;
    int l   = 2 * lp + par;
    const float* U; const float* W;
    int mloc, i, j;
    if (r < 256) { U = U3_0; W = W3_0; mloc = 0;        i = (r >> 4) & 15; j = r & 15; }
    else { int rl = r - 256; U = U3_1; W = W3_1; mloc = rl >> 8; i = (rl >> 4) & 15; j = rl & 15; }
    const float* u = U + ((((mloc * NI + i) * NI + j) * NI + l) * 4);     // K3=4 fastest
    const float* w = W + (e * 4) * NCHAN + c;                             // W3[e,k,c]
    Ueff3[idx] = u[0] * w[0] + u[1] * w[NCHAN] + u[2] * w[2 * NCHAN] + u[3] * w[3 * NCHAN];
}

// ---------------------------------------------------------------------------
// Kernel A2: Ueff2[e][c][r] = sum_k U2_t[m,i,j,k] * W2_t[e,k,c]
// ---------------------------------------------------------------------------
__global__ void build_ueff2(const float* __restrict__ U2_0, const float* __restrict__ W2_0,
                            const float* __restrict__ U2_1, const float* __restrict__ W2_1,
                            float* __restrict__ Ueff2) {
    int idx = blockIdx.x * blockDim.x + threadIdx.x;
    if (idx >= NELEM * NCHAN * RTOT) return;
    int r  = idx & (RTOT - 1);
    int ec = idx >> 10;
    int c  = ec & 63;
    int e  = ec >> 6;
    const float* U; const float* W;
    int mloc, i, j;
    if (r < 256) { U = U2_0; W = W2_0; mloc = 0;        i = (r >> 4) & 15; j = r & 15; }
    else { int rl = r - 256; U = U2_1; W = W2_1; mloc = rl >> 8; i = (rl >> 4) & 15; j = rl & 15; }
    const float* u = U + (((mloc * NI + i) * NI + j) * 2);                // K2=2 fastest
    const float* w = W + (e * 2) * NCHAN + c;
    Ueff2[idx] = u[0] * w[0] + u[1] * w[NCHAN];
}

// ---------------------------------------------------------------------------
// Kernel A1: Ueff1[e][c][m*16+i] = U1_t[m,i,0] * W1_t[e,0,c]   (K1=1)
// ---------------------------------------------------------------------------
__global__ void build_ueff1(const float* __restrict__ U1_0, const float* __restrict__ W1_0,
                            const float* __restrict__ U1_1, const float* __restrict__ W1_1,
                            float* __restrict__ Ueff1) {
    int idx = blockIdx.x * blockDim.x + threadIdx.x;
    if (idx >= NELEM * NCHAN * 64) return;
    int mi = idx & 63;             // m*16 + i
    int ec = idx >> 6;
    int c  = ec & 63;
    int e  = ec >> 6;
    int m  = mi >> 4;
    int i  = mi & 15;
    const float* U; const float* W; int mloc;
    if (m == 0) { U = U1_0; W = W1_0; mloc = 0; }
    else        { U = U1_1; W = W1_1; mloc = m - 1; }
    Ueff1[idx] = U[mloc * NI + i] * W[e * NCHAN + c];
}

// ---------------------------------------------------------------------------
// Kernel E1: elem[b] = index of the one-hot element in node_attrs
// ---------------------------------------------------------------------------
__global__ void find_elem(const float* __restrict__ y, int* __restrict__ elem) {
    int b = blockIdx.x * blockDim.x + threadIdx.x;
    if (b >= NNODES) return;
    int e = 0;
    #pragma unroll
    for (int k = 0; k < NELEM; ++k)
        if (y[b * NELEM + k] > 0.5f) e = k;
    elem[b] = e;
}

// ---------------------------------------------------------------------------
// Kernel E2: deterministic stable grouping of nodes by element into 16-node
// tiles: slotNode[tile*16+s], slotMask (valid flag), tileElem[tile] (-1 = skip)
// ---------------------------------------------------------------------------
__global__ void build_groups(const int* __restrict__ elem,
                             int* __restrict__ slotNode, int* __restrict__ slotMask,
                             int* __restrict__ tileElem) {
    __shared__ int elemS[NNODES];
    __shared__ int cntS[NELEM];
    __shared__ int tbaseS[NELEM];
    int tid = threadIdx.x;                  // 256 threads
    for (int b = tid; b < NNODES; b += 256) elemS[b] = elem[b];
    if (tid < NELEM) cntS[tid] = 0;
    for (int t = tid; t < MAX_TILES; t += 256) tileElem[t] = -1;
    for (int s = tid; s < MAX_TILES * 16; s += 256) { slotNode[s] = 0; slotMask[s] = 0; }
    __syncthreads();
    for (int b = tid; b < NNODES; b += 256) atomicAdd(&cntS[elemS[b]], 1);
    __syncthreads();
    if (tid == 0) {
        int tb = 0;
        for (int e = 0; e < NELEM; ++e) {
            tbaseS[e] = tb;
            int nt = (cntS[e] + 15) >> 4;
            for (int k = 0; k < nt; ++k) tileElem[tb + k] = e;
            tb += nt;
        }
    }
    __syncthreads();
    // stable scatter: rank of b within its element class
    for (int n = 0; n < 8; ++n) {
        int b = tid * 8 + n;
        int eb = elemS[b];
        int rank = 0;
        for (int bp = 0; bp < b; ++bp) rank += (elemS[bp] == eb) ? 1 : 0;
        int slot = tbaseS[eb] * 16 + rank;
        slotNode[slot] = b;
        slotMask[slot] = 1;
    }
}

// ---------------------------------------------------------------------------
// Main kernel: per (16-node tile, channel): WMMA contraction
//   D[r,b] = sum_l Ueff3[e,c,l,r] * x[b,c,l]  + Ueff2[e,c,r]   (C operand)
// then per-node fold:  out2 = sum_j D, c1 = out2 + Ueff1, f_m += c1 * x_i
// 8 waves / block, 1 channel per wave.  All lane-half-dependent x operands are
// loaded directly from memory (no register-array selects -> no cndmask chains).
// ---------------------------------------------------------------------------
__global__ void __launch_bounds__(256)
mace_contract(const float* __restrict__ x,
              const float* __restrict__ Ueff3, const float* __restrict__ Ueff2,
              const float* __restrict__ Ueff1,
              const int* __restrict__ slotNode, const int* __restrict__ slotMask,
              const int* __restrict__ tileElem,
              float* __restrict__ f0, float* __restrict__ f1) {
    int tile = blockIdx.x;
    int e = tileElem[tile];
    if (e < 0) return;                      // uniform across block: safe for WMMA

    int wave = threadIdx.x >> 5;
    int lane = threadIdx.x & 31;
    int col  = lane & 15;                   // node column within tile
    int hi   = lane >> 4;                   // lane half (K/row split)
    int c    = blockIdx.y * 8 + wave;       // channel handled by this wave

    int slot  = tile * 16 + col;
    int b     = slotNode[slot];             // padded slots point at node 0 (valid mem)
    int valid = slotMask[slot];

    const float* xb = x + ((size_t)b * NCHAN + c) * NI;   // this node's 16 x values

    // B operands (4x16 f32 layout: vgpr0 = K0 (lanes 0-15)/K2 (16-31), vgpr1 = K1/K3):
    // lane needs x[4q + 2*hi], x[4q + 2*hi + 1]  -> one b64 load each, no selects.
    v2f B0 = *(const v2f*)(xb + 0  + 2 * hi);
    v2f B1 = *(const v2f*)(xb + 4  + 2 * hi);
    v2f B2 = *(const v2f*)(xb + 8  + 2 * hi);
    v2f B3 = *(const v2f*)(xb + 12 + 2 * hi);

    // j-contraction operand: lane half owns j = v + 8*hi -> two b128 loads.
    float4 xs0 = *(const float4*)(xb + 8 * hi);
    float4 xs1 = *(const float4*)(xb + 8 * hi + 4);

    // full x row, used only with compile-time indices (i-contraction).
    float4 a0 = ((const float4*)xb)[0], a1 = ((const float4*)xb)[1];
    float4 a2 = ((const float4*)xb)[2], a3 = ((const float4*)xb)[3];
    float xr[16] = {a0.x, a0.y, a0.z, a0.w, a1.x, a1.y, a1.z, a1.w,
                    a2.x, a2.y, a2.z, a2.w, a3.x, a3.y, a3.z, a3.w};

    const float* U3c = Ueff3 + (size_t)(e * NCHAN + c) * NI * RTOT;  // [lp][r][par]
    const float* U2p = Ueff2 + (size_t)(e * NCHAN + c) * RTOT + 8 * hi;
    const float* U1  = Ueff1 + (size_t)(e * NCHAN + c) * 64;         // [m*16+i]

    for (int m = 0; m < 4; ++m) {           // not unrolled: caps register pressure
        float facc = 0.f;
        #pragma unroll
        for (int i = 0; i < 16; ++i) {
            const int r0 = (m * 16 + i) * 16;   // = m*256 + i*16

            // A operands: interleaved layout -> one b64 per chunk (lp = 2q + hi)
            v2f A0 = *(const v2f*)(U3c + (size_t)(0 + hi) * (2 * RTOT) + 2 * (r0 + col));
            v2f A1 = *(const v2f*)(U3c + (size_t)(2 + hi) * (2 * RTOT) + 2 * (r0 + col));
            v2f A2 = *(const v2f*)(U3c + (size_t)(4 + hi) * (2 * RTOT) + 2 * (r0 + col));
            v2f A3 = *(const v2f*)(U3c + (size_t)(6 + hi) * (2 * RTOT) + 2 * (r0 + col));

            // C operand = Ueff2 tile rows (row = v + 8*hi): two b128 loads
            float4 cL = *(const float4*)(U2p + r0);
            float4 cH = *(const float4*)(U2p + r0 + 4);

            v8f acc;
            acc[0] = cL.x; acc[1] = cL.y; acc[2] = cL.z; acc[3] = cL.w;
            acc[4] = cH.x; acc[5] = cH.y; acc[6] = cH.z; acc[7] = cH.w;

            acc = __builtin_amdgcn_wmma_f32_16x16x4_f32(false, A0, false, B0, (short)0, acc, false, false);
            acc = __builtin_amdgcn_wmma_f32_16x16x4_f32(false, A1, false, B1, (short)0, acc, false, false);
            acc = __builtin_amdgcn_wmma_f32_16x16x4_f32(false, A2, false, B2, (short)0, acc, false, false);
            acc = __builtin_amdgcn_wmma_f32_16x16x4_f32(false, A3, false, B3, (short)0, acc, false, false);

            // out2 = sum_j c2[j] * x[j]; this lane half owns j = v + 8*hi
            float part = acc[0] * xs0.x + acc[1] * xs0.y + acc[2] * xs0.z + acc[3] * xs0.w
                       + acc[4] * xs1.x + acc[5] * xs1.y + acc[6] * xs1.z + acc[7] * xs1.w;
            part += __shfl_xor(part, 16, 32);

            float c1v = part + U1[m * 16 + i];
            facc += c1v * xr[i];
        }
        if (valid && hi == 0) {
            if (m == 0) f0[b * NCHAN + c] = facc;
            else        f1[((size_t)b * NCHAN + c) * 3 + (m - 1)] = facc;
        }
    }
}

// ---------------------------------------------------------------------------
// Epilogue: channel-mixing linear + inv_sqrt(C) + skip connection
// out[b, f]          = sum_c f0[b,c] Wlin0[c,f] / 8 + sc[b,f]           (f<64)
// out[b, 64+f*3+m]   = sum_c f1[b,c,m] Wlin1[c,f] / 8 + sc[b,64+f*3+m]
// ---------------------------------------------------------------------------
__global__ void mace_linear(const float* __restrict__ f0, const float* __restrict__ f1,
                            const float* __restrict__ Wlin0, const float* __restrict__ Wlin1,
                            const float* __restrict__ sc, float* __restrict__ out) {
    int idx = blockIdx.x * blockDim.x + threadIdx.x;
    if (idx >= NNODES * 256) return;
    int b = idx >> 8, colo = idx & 255;
    float s = 0.f;
    if (colo < 64) {
        int f = colo;
        const float* fr = f0 + b * NCHAN;
        #pragma unroll 8
        for (int c = 0; c < NCHAN; ++c) s += fr[c] * Wlin0[c * NCHAN + f];
    } else {
        int q = colo - 64;
        int f = q / 3, m = q - 3 * f;
        const float* fr = f1 + (size_t)b * NCHAN * 3 + m;
        #pragma unroll 8
        for (int c = 0; c < NCHAN; ++c) s += fr[c * 3] * Wlin1[c * NCHAN + f];
    }
    out[idx] = s * 0.125f + sc[idx];
}

// ---------------------------------------------------------------------------
extern "C" void kernel_launch(void* const* d_in, const int* in_sizes, int n_in,
                              void* d_out, int out_size, void* d_ws, size_t ws_size,
                              hipStream_t stream) {
    const float* x    = (const float*)d_in[0];
    const float* sc   = (const float*)d_in[1];
    const float* y    = (const float*)d_in[2];
    const float* U3_0 = (const float*)d_in[3];
    const float* U2_0 = (const float*)d_in[4];
    const float* U1_0 = (const float*)d_in[5];
    const float* W3_0 = (const float*)d_in[6];
    const float* W2_0 = (const float*)d_in[7];
    const float* W1_0 = (const float*)d_in[8];
    const float* Wl_0 = (const float*)d_in[9];
    const float* U3_1 = (const float*)d_in[10];
    const float* U2_1 = (const float*)d_in[11];
    const float* U1_1 = (const float*)d_in[12];
    const float* W3_1 = (const float*)d_in[13];
    const float* W2_1 = (const float*)d_in[14];
    const float* W1_1 = (const float*)d_in[15];
    const float* Wl_1 = (const float*)d_in[16];

    // workspace layout (floats then ints), ~46.9 MB total
    float* Ueff3 = (float*)d_ws;                         // 10*64*16*1024
    float* Ueff2 = Ueff3 + (size_t)NELEM * NCHAN * NI * RTOT;
    float* Ueff1 = Ueff2 + (size_t)NELEM * NCHAN * RTOT;
    float* f0    = Ueff1 + (size_t)NELEM * NCHAN * 64;
    float* f1    = f0 + (size_t)NNODES * NCHAN;
    int*   elem     = (int*)(f1 + (size_t)NNODES * NCHAN * 3);
    int*   slotNode = elem + NNODES;
    int*   slotMask = slotNode + MAX_TILES * 16;
    int*   tileElem = slotMask + MAX_TILES * 16;

    {
        int n = NELEM * NCHAN * NI * RTOT;
        build_ueff3<<<(n + 255) / 256, 256, 0, stream>>>(U3_0, W3_0, U3_1, W3_1, Ueff3);
    }
    {
        int n = NELEM * NCHAN * RTOT;
        build_ueff2<<<(n + 255) / 256, 256, 0, stream>>>(U2_0, W2_0, U2_1, W2_1, Ueff2);
    }
    {
        int n = NELEM * NCHAN * 64;
        build_ueff1<<<(n + 255) / 256, 256, 0, stream>>>(U1_0, W1_0, U1_1, W1_1, Ueff1);
    }
    find_elem<<<(NNODES + 255) / 256, 256, 0, stream>>>(y, elem);
    build_groups<<<1, 256, 0, stream>>>(elem, slotNode, slotMask, tileElem);

    dim3 grid(MAX_TILES, 8);   // 8 channel-groups of 8 waves x 1 channel
    mace_contract<<<grid, 256, 0, stream>>>(x, Ueff3, Ueff2, Ueff1,
                                            slotNode, slotMask, tileElem, f0, f1);

    {
        int n = NNODES * 256;
        mace_linear<<<(n + 255) / 256, 256, 0, stream>>>(f0, f1, Wl_0, Wl_1, sc, (float*)d_out);
    }
}